// MeanDegConv_49658411876806
// MI455X (gfx1250) — compile-verified
//
#include <hip/hip_runtime.h>
#include <math.h>

// MeanDegConv on MI455X (gfx1250): fp32 WMMA (V_WMMA_F32_16X16X4_F32) GEMMs +
// TDM (tensor_load_to_lds) async tile staging + L2-resident float atomics.
#define DV   128
#define NV   50000
#define NE   10000
#define NNZT 1000000
#define LDA  132   // padded LDS row stride (floats): 128 + 4 (TDM pad_amount=3)

typedef __attribute__((ext_vector_type(2))) float v2f;
typedef __attribute__((ext_vector_type(8))) float v8f;
typedef __attribute__((ext_vector_type(4))) unsigned int u32x4;
typedef __attribute__((ext_vector_type(8))) int i32x8;
typedef __attribute__((ext_vector_type(4))) int i32x4;

#if defined(__gfx1250__) && __has_builtin(__builtin_amdgcn_tensor_load_to_lds) && \
    __has_builtin(__builtin_amdgcn_s_wait_tensorcnt)
#define USE_TDM 1
#else
#define USE_TDM 0
#endif

// Stage one 16x128 f32 tile (row stride DV) from global into LDS (row stride LDA).
// TDM path: single DMA descriptor, pad_interval=6 (128 DWORDs) + pad_amount=3
// (4 DWORDs) reproduces the LDA=132 padded layout. Issued by wave 0 only.
__device__ __forceinline__ void tile_issue(const float* __restrict__ gsrc,
                                           float* lds_dst) {
#if USE_TDM
  if (threadIdx.x < 32) {
    const unsigned lds_off = (unsigned)(uintptr_t)lds_dst;   // flat addr[31:0] == LDS byte addr
    const unsigned long long ga = (unsigned long long)(uintptr_t)gsrc;
    u32x4 g0;
    g0[0] = 1u;                                          // count=1, user descriptor
    g0[1] = lds_off;                                     // lds_addr (bytes)
    g0[2] = (unsigned)(ga & 0xFFFFFFFFu);                // global_addr[31:0]
    g0[3] = (unsigned)((ga >> 32) & 0x01FFFFFFu)         // global_addr[56:32]
            | (2u << 30);                                // type = 2 ("image")
    i32x8 g1;
    g1[0] = (int)((2u << 16)      // data_size = 4 bytes
                  | (1u << 20)    // pad_enable
                  | (6u << 22)    // pad_interval: 128 DWORDs
                  | (3u << 25));  // pad_amount:   4 DWORDs
    g1[1] = (int)(128u << 16);    // tensor_dim0 = 128 (bits 79:48, low half)
    g1[2] = (int)(16u << 16);     // tensor_dim0 hi=0 ; tensor_dim1 = 16 (bits 95:80)
    g1[3] = (int)(128u << 16);    // tensor_dim1 hi=0 ; tile_dim0 = 128
    g1[4] = 16;                   // tile_dim1 = 16 ; tile_dim2 = 0
    g1[5] = 128;                  // tensor_dim0_stride = 128 elements
    g1[6] = 0;
    g1[7] = 0;
    const i32x4 z4 = {0, 0, 0, 0};
    const i32x8 z8 = {0, 0, 0, 0, 0, 0, 0, 0};
    __builtin_amdgcn_tensor_load_to_lds(g0, g1, z4, z4, z8, 0);
  }
#else
  for (int t = threadIdx.x; t < 16 * DV; t += blockDim.x) {
    int r = t >> 7, cix = t & 127;
    lds_dst[r * LDA + cix] = gsrc[(size_t)r * DV + cix];
  }
#endif
}

__device__ __forceinline__ void tile_wait() {
#if USE_TDM
  if (threadIdx.x < 32) __builtin_amdgcn_s_wait_tensorcnt(0);
#endif
}

// One 16x16 output tile, K=128, A staged in LDS (16 x LDA), B row-major [128 x N=128].
// f32 WMMA fragment layout (ISA 7.12.2):
//  A 16x4 : lane L holds M=L%16, K = {0,1} (lanes 0-15) or {2,3} (lanes 16-31)
//  B 4x16 : lane L holds N=L%16, same K-pair striping
//  C/D    : vgpr r: lanes0-15 -> M=r, lanes16-31 -> M=r+8, N=lane%16
__device__ __forceinline__ v8f wmma_k128(v8f c, const float* __restrict__ sa,
                                         const float* __restrict__ B,
                                         int n0, int lane) {
  const int row   = lane & 15;
  const int khalf = (lane >> 4) << 1;          // 0 or 2
  const int col   = n0 + row;
#pragma unroll
  for (int k = 0; k < DV; k += 4) {
    const int kb = k + khalf;
    v2f a;
    a.x = sa[row * LDA + kb];
    a.y = sa[row * LDA + kb + 1];
    v2f b;
    b.x = B[(size_t)kb * DV + col];
    b.y = B[(size_t)(kb + 1) * DV + col];
    c = __builtin_amdgcn_wmma_f32_16x16x4_f32(false, a, false, b, (short)0, c,
                                              false, false);
  }
  return c;
}

__global__ void k_zero(float* __restrict__ p, size_t n) {
  size_t i = (size_t)blockIdx.x * blockDim.x + threadIdx.x;
  size_t s = (size_t)gridDim.x * blockDim.x;
  for (; i < n; i += s) p[i] = 0.0f;
}

__global__ void k_deg(const int* __restrict__ vertex, const int* __restrict__ edges,
                      float* __restrict__ deg_v, float* __restrict__ deg_e) {
  int i = blockIdx.x * 256 + threadIdx.x;
  if (i < NNZT) {
    atomicAdd(&deg_e[edges[i]], 1.0f);
    atomicAdd(&deg_v[vertex[i]], 1.0f);
  }
}

// Y1 = X @ W1 + b1 ; Y2 = X @ W2[0:128,:]   (16 rows per block, 8 waves x 16 cols)
__global__ void k_gemm_xw(const float* __restrict__ X, const float* __restrict__ W1,
                          const float* __restrict__ b1, const float* __restrict__ W2,
                          float* __restrict__ Y1, float* __restrict__ Y2) {
  __shared__ float sA[16 * LDA];
  const int m0 = blockIdx.x * 16;
  tile_issue(X + (size_t)m0 * DV, sA);
  tile_wait();
  __syncthreads();
  const int lane = threadIdx.x & 31;
  const int n0 = (threadIdx.x >> 5) * 16;
  v8f c1 = {0.f, 0.f, 0.f, 0.f, 0.f, 0.f, 0.f, 0.f};
  v8f c2 = c1;
  c1 = wmma_k128(c1, sA, W1, n0, lane);
  c2 = wmma_k128(c2, sA, W2, n0, lane);
  const int n = n0 + (lane & 15);
  const int mb = m0 + ((lane >> 4) << 3);
  const float bias = b1[n];
#pragma unroll
  for (int r = 0; r < 8; ++r) {
    Y1[(size_t)(mb + r) * DV + n] = c1[r] + bias;
    Y2[(size_t)(mb + r) * DV + n] = c2[r];
  }
}

// One wave per nnz: Eacc[edges[i]] += Y1[vertex[i]]  (float4 load, 4 atomics/lane)
__global__ void k_scatter_e(const float* __restrict__ Y1, const int* __restrict__ vertex,
                            const int* __restrict__ edges, float* __restrict__ Eacc) {
  const long long i = (long long)blockIdx.x * 8 + (threadIdx.x >> 5);
  if (i >= NNZT) return;
  const int lane = threadIdx.x & 31;
  const int v = vertex[i];
  const int e = edges[i];
  const float4 y = ((const float4*)(Y1 + (size_t)v * DV))[lane];
  float* dst = Eacc + (size_t)e * DV + lane * 4;
  atomicAdd(dst + 0, y.x);
  atomicAdd(dst + 1, y.y);
  atomicAdd(dst + 2, y.z);
  atomicAdd(dst + 3, y.w);
}

// Xe = Eacc / deg_e (in place), logde = log(deg_e)
__global__ void k_edge_mean(float* __restrict__ Eacc, const float* __restrict__ deg_e,
                            float* __restrict__ logde) {
  const int i = blockIdx.x * 256 + threadIdx.x;
  if (i < NE * DV) Eacc[i] = Eacc[i] / deg_e[i >> 7];
  if (i < NE) logde[i] = logf(deg_e[i]);
}

// Z2 = Xe @ W2[128:256,:] + logde * W2[256,:] + W2_b
__global__ void k_gemm_z2(const float* __restrict__ Xe, const float* __restrict__ logde,
                          const float* __restrict__ W2, const float* __restrict__ W2b,
                          float* __restrict__ Z2) {
  __shared__ float sA[16 * LDA];
  const int m0 = blockIdx.x * 16;
  tile_issue(Xe + (size_t)m0 * DV, sA);
  tile_wait();
  __syncthreads();
  const int lane = threadIdx.x & 31;
  const int n0 = (threadIdx.x >> 5) * 16;
  v8f c = {0.f, 0.f, 0.f, 0.f, 0.f, 0.f, 0.f, 0.f};
  c = wmma_k128(c, sA, W2 + (size_t)128 * DV, n0, lane);
  const int n = n0 + (lane & 15);
  const int mb = m0 + ((lane >> 4) << 3);
  const float wlog = W2[(size_t)256 * DV + n];
  const float bias = W2b[n];
#pragma unroll
  for (int r = 0; r < 8; ++r)
    Z2[(size_t)(mb + r) * DV + n] = c[r] + logde[mb + r] * wlog + bias;
}

// One wave per nnz: Vacc[vertex[i]] += Z2[edges[i]]
__global__ void k_scatter_v(const float* __restrict__ Z2, const int* __restrict__ vertex,
                            const int* __restrict__ edges, float* __restrict__ Vacc) {
  const long long i = (long long)blockIdx.x * 8 + (threadIdx.x >> 5);
  if (i >= NNZT) return;
  const int lane = threadIdx.x & 31;
  const int v = vertex[i];
  const int e = edges[i];
  const float4 z = ((const float4*)(Z2 + (size_t)e * DV))[lane];
  float* dst = Vacc + (size_t)v * DV + lane * 4;
  atomicAdd(dst + 0, z.x);
  atomicAdd(dst + 1, z.y);
  atomicAdd(dst + 2, z.z);
  atomicAdd(dst + 3, z.w);
}

// Xv = Y2 + Vacc/deg_v ; h = [Xv, X, X0, log(deg_v)] ;
// out = relu(h @ W3_1 + b3_1) @ W3_2 + b3_2
__global__ void k_final(const float* __restrict__ Y2, const float* __restrict__ Vacc,
                        const float* __restrict__ deg_v, const float* __restrict__ X,
                        const float* __restrict__ X0, const float* __restrict__ W31,
                        const float* __restrict__ b31, const float* __restrict__ W32,
                        const float* __restrict__ b32, float* __restrict__ out) {
  __shared__ float sXv[16 * LDA];
  __shared__ float sX[16 * LDA];
  __shared__ float sX0[16 * LDA];
  __shared__ float sT[16 * LDA];
  __shared__ float sLd[16];
  const int m0 = blockIdx.x * 16;
  // Kick off the X / X0 tile DMAs, overlap them with the Xv computation.
  tile_issue(X + (size_t)m0 * DV, sX);
  tile_issue(X0 + (size_t)m0 * DV, sX0);
  for (int t = threadIdx.x; t < 16 * DV; t += blockDim.x) {
    int r = t >> 7, cix = t & 127;
    size_t g = (size_t)(m0 + r) * DV + cix;
    sXv[r * LDA + cix] = Y2[g] + Vacc[g] / deg_v[m0 + r];
  }
  if (threadIdx.x < 16) sLd[threadIdx.x] = logf(deg_v[m0 + threadIdx.x]);
  tile_wait();
  __syncthreads();
  const int lane = threadIdx.x & 31;
  const int n0 = (threadIdx.x >> 5) * 16;
  v8f c = {0.f, 0.f, 0.f, 0.f, 0.f, 0.f, 0.f, 0.f};
  c = wmma_k128(c, sXv, W31, n0, lane);                       // rows   0..127
  c = wmma_k128(c, sX, W31 + (size_t)128 * DV, n0, lane);     // rows 128..255
  c = wmma_k128(c, sX0, W31 + (size_t)256 * DV, n0, lane);    // rows 256..383
  const int n = n0 + (lane & 15);
  const int mb = (lane >> 4) << 3;
  const float wl = W31[(size_t)384 * DV + n];
  const float bb = b31[n];
#pragma unroll
  for (int r = 0; r < 8; ++r) {
    float vv = c[r] + sLd[mb + r] * wl + bb;
    sT[(mb + r) * LDA + n] = fmaxf(vv, 0.0f);
  }
  __syncthreads();
  v8f o = {0.f, 0.f, 0.f, 0.f, 0.f, 0.f, 0.f, 0.f};
  o = wmma_k128(o, sT, W32, n0, lane);
  const float b2v = b32[n];
#pragma unroll
  for (int r = 0; r < 8; ++r)
    out[(size_t)(m0 + mb + r) * DV + n] = o[r] + b2v;
}

extern "C" void kernel_launch(void* const* d_in, const int* in_sizes, int n_in,
                              void* d_out, int out_size, void* d_ws, size_t ws_size,
                              hipStream_t stream) {
  const float* X = (const float*)d_in[0];
  const float* X0 = (const float*)d_in[1];
  const int* vertex = (const int*)d_in[2];
  const int* edges = (const int*)d_in[3];
  const float* W1w = (const float*)d_in[4];
  const float* W1b = (const float*)d_in[5];
  const float* W2w = (const float*)d_in[6];
  const float* W2b = (const float*)d_in[7];
  const float* W31 = (const float*)d_in[8];
  const float* b31 = (const float*)d_in[9];
  const float* W32 = (const float*)d_in[10];
  const float* b32 = (const float*)d_in[11];

  float* ws = (float*)d_ws;
  float* Eacc = ws;                           // [E*D]  zeroed
  float* Vacc = Eacc + (size_t)NE * DV;       // [N*D]  zeroed
  float* deg_e = Vacc + (size_t)NV * DV;      // [E]    zeroed
  float* deg_v = deg_e + NE;                  // [N]    zeroed
  float* Y1 = deg_v + NV;                     // [N*D]
  float* Y2 = Y1 + (size_t)NV * DV;           // [N*D]
  float* Z2 = Y2 + (size_t)NV * DV;           // [E*D]
  float* logde = Z2 + (size_t)NE * DV;        // [E]

  const size_t zcount = (size_t)NE * DV + (size_t)NV * DV + NE + NV;
  k_zero<<<2048, 256, 0, stream>>>(ws, zcount);
  k_deg<<<(NNZT + 255) / 256, 256, 0, stream>>>(vertex, edges, deg_v, deg_e);
  k_gemm_xw<<<NV / 16, 256, 0, stream>>>(X, W1w, W1b, W2w, Y1, Y2);
  k_scatter_e<<<NNZT / 8, 256, 0, stream>>>(Y1, vertex, edges, Eacc);
  k_edge_mean<<<(NE * DV + 255) / 256, 256, 0, stream>>>(Eacc, deg_e, logde);
  k_gemm_z2<<<NE / 16, 256, 0, stream>>>(Eacc, logde, W2w, W2b, Z2);
  k_scatter_v<<<NNZT / 8, 256, 0, stream>>>(Z2, vertex, edges, Vacc);
  k_final<<<NV / 16, 256, 0, stream>>>(Y2, Vacc, deg_v, X, X0, W31, b31, W32, b32,
                                       (float*)d_out);
}